// FarthestPointsDecimate_29472065585600
// MI455X (gfx1250) — compile-verified
//
#include <hip/hip_runtime.h>
#include <hip/hip_bf16.h>
#include <cfloat>
#include <cstdint>

// ---------------------------------------------------------------------------
// Batched farthest-point sampling (FPS), MI455X / gfx1250.
//
//   coords: (L=16384, N=32, C=3) f32.  K = ceil(0.25*L) = 4096.
//   out   : keep (K*N int32, layout out[k*N+n]) then batch (K*N int32, =n).
//
// One workgroup per cloud; the whole cloud (192 KB) lives in the WGP's
// 320 KB LDS, staged once with gfx1250 async global->LDS loads (ASYNCcnt).
// Each thread caches its 16 points + running min-dist in VGPRs; per step we
// do a fused min-update + argmax with wave32 shuffle reduction and a tiny
// cross-wave LDS reduction.  HBM is touched exactly once (6 MB in, 1 MB out).
// ---------------------------------------------------------------------------

#define L_POINTS 16384
#define N_BATCH  32
#define K_KEEP   4096              // ceil(0.25 * 16384)
#define BLOCK    1024              // 32 wave32's -> fills one WGP
#define PPT      (L_POINTS / BLOCK) // 16 points per thread

// gfx1250 async global->LDS DMA (tracked by ASYNCcnt).  This toolchain
// declares the builtins with generic `int*` parameters (per its diagnostics),
// so we pass plain int* casts; the backend infers address spaces from
// pointer provenance (kernarg global pointer / __shared__ array).
#if defined(__has_builtin)
#if __has_builtin(__builtin_amdgcn_global_load_async_to_lds_b32) && \
    __has_builtin(__builtin_amdgcn_s_wait_asynccnt)
#define USE_ASYNC_LDS 1
#endif
#endif
#ifndef USE_ASYNC_LDS
#define USE_ASYNC_LDS 0
#endif

extern "C" __global__ __launch_bounds__(BLOCK, 1)
void fps_kernel(const float* __restrict__ coords,
                int* __restrict__ keep_out,
                int* __restrict__ batch_out)
{
    // Dynamic LDS: xs/ys/zs SoA (192 KB) + reduction scratch + broadcast slot.
    extern __shared__ float smem[];
    float* xs   = smem;                      // [L_POINTS]
    float* ys   = xs + L_POINTS;             // [L_POINTS]
    float* zs   = ys + L_POINTS;             // [L_POINTS]
    float* wval = zs + L_POINTS;             // [32] per-wave best value
    int*   widx = (int*)(wval + 32);         // [32] per-wave best index
    float* bq   = (float*)(widx + 32);       // [3]  broadcast query point

    const int n    = blockIdx.x;             // cloud id (0..31)
    const int tid  = threadIdx.x;
    const int lane = tid & 31;
    const int wid  = tid >> 5;

    // ---- Stage cloud n from (L,N,3) global layout into LDS SoA -----------
    // element (l,n,c) at flat float index l*(3*N) + 3*n + c  (4B aligned).
    const float* gbase = coords + 3 * n;
#if USE_ASYNC_LDS
    #pragma unroll
    for (int j = 0; j < PPT; ++j) {
        const int l = tid + j * BLOCK;
        const float* g = gbase + (size_t)l * (3 * N_BATCH);
        __builtin_amdgcn_global_load_async_to_lds_b32((int*)(g + 0), (int*)&xs[l], 0, 0);
        __builtin_amdgcn_global_load_async_to_lds_b32((int*)(g + 1), (int*)&ys[l], 0, 0);
        __builtin_amdgcn_global_load_async_to_lds_b32((int*)(g + 2), (int*)&zs[l], 0, 0);
    }
    __builtin_amdgcn_s_wait_asynccnt(0);     // my wave's DMA done before barrier
#else
    #pragma unroll
    for (int j = 0; j < PPT; ++j) {
        const int l = tid + j * BLOCK;
        const float* g = gbase + (size_t)l * (3 * N_BATCH);
        xs[l] = g[0]; ys[l] = g[1]; zs[l] = g[2];
    }
#endif
    __syncthreads();

    // ---- Cache owned points + min-dist in VGPRs ---------------------------
    float px[PPT], py[PPT], pz[PPT], md[PPT];
    #pragma unroll
    for (int j = 0; j < PPT; ++j) {
        const int l = tid + j * BLOCK;
        px[j] = xs[l]; py[j] = ys[l]; pz[j] = zs[l];
        md[j] = FLT_MAX;   // first pass turns this into dist-to-point-0
    }

    // Deterministic start: point 0 is the first selection.
    float qx = xs[0], qy = ys[0], qz = zs[0];
    if (tid == 0) {
        keep_out[n]  = 0;   // keep[0*N + n]
        batch_out[n] = n;   // batch[0*N + n]
    }

    // ---- K-1 sequential FPS steps ----------------------------------------
    for (int k = 1; k < K_KEEP; ++k) {
        // fused: md = min(md, dist(p, q));  argmax over updated md
        float bV = -FLT_MAX;
        int   bI = 0x7fffffff;
        #pragma unroll
        for (int j = 0; j < PPT; ++j) {
            const float dx = px[j] - qx;
            const float dy = py[j] - qy;
            const float dz = pz[j] - qz;
            const float d  = fmaf(dx, dx, fmaf(dy, dy, dz * dz));
            const float m  = fminf(md[j], d);
            md[j] = m;
            if (m > bV) { bV = m; bI = tid + j * BLOCK; }  // strict > : lowest idx wins
        }

        // wave32 argmax reduction (lowest index on value ties, like jnp.argmax)
        #pragma unroll
        for (int off = 16; off > 0; off >>= 1) {
            const float oV = __shfl_xor(bV, off, 32);
            const int   oI = __shfl_xor(bI, off, 32);
            if (oV > bV || (oV == bV && oI < bI)) { bV = oV; bI = oI; }
        }
        if (lane == 0) { wval[wid] = bV; widx[wid] = bI; }
        __syncthreads();

        // cross-wave reduction by wave 0 (32 entries = 1 wave)
        if (wid == 0) {
            float v = wval[lane];
            int   i = widx[lane];
            #pragma unroll
            for (int off = 16; off > 0; off >>= 1) {
                const float oV = __shfl_xor(v, off, 32);
                const int   oI = __shfl_xor(i, off, 32);
                if (oV > v || (oV == v && oI < i)) { v = oV; i = oI; }
            }
            if (lane == 0) {
                bq[0] = xs[i]; bq[1] = ys[i]; bq[2] = zs[i];
                keep_out[(size_t)k * N_BATCH + n]  = i;  // batch-local index
                batch_out[(size_t)k * N_BATCH + n] = n;
            }
        }
        __syncthreads();
        qx = bq[0]; qy = bq[1]; qz = bq[2];
    }
}

// ---------------------------------------------------------------------------
extern "C" void kernel_launch(void* const* d_in, const int* in_sizes, int n_in,
                              void* d_out, int out_size, void* d_ws, size_t ws_size,
                              hipStream_t stream)
{
    (void)in_sizes; (void)n_in; (void)out_size; (void)d_ws; (void)ws_size;

    const float* coords = (const float*)d_in[0];
    // Reference returns int32 (keep, batch) concatenated: 2 * K * N elements.
    int* keep_out  = (int*)d_out;
    int* batch_out = keep_out + (size_t)K_KEEP * N_BATCH;

    // 192 KB coords + reduction/broadcast scratch (>64 KB default -> opt in).
    const size_t lds_bytes =
        (size_t)(3 * L_POINTS + 32 + 32 + 4) * sizeof(float);
    (void)hipFuncSetAttribute((const void*)fps_kernel,
                              hipFuncAttributeMaxDynamicSharedMemorySize,
                              (int)lds_bytes);

    fps_kernel<<<dim3(N_BATCH), dim3(BLOCK), lds_bytes, stream>>>(
        coords, keep_out, batch_out);
}